// KGVAE_90426241450702
// MI455X (gfx1250) — compile-verified
//
#include <hip/hip_runtime.h>

typedef __attribute__((ext_vector_type(2))) float v2f;
typedef __attribute__((ext_vector_type(8))) float v8f;
typedef int v4i __attribute__((vector_size(16)));   // builtin's pointee type

// CDNA5 async memory->LDS path (GLOBAL_LOAD_ASYNC_TO_LDS_B128, ASYNCcnt).
#if defined(__has_builtin)
#if __has_builtin(__builtin_amdgcn_global_load_async_to_lds_b128) && \
    __has_builtin(__builtin_amdgcn_s_wait_asynccnt)
#define USE_ASYNC_LDS 1
#endif
#endif

#define AS1 __attribute__((address_space(1)))
#define AS3 __attribute__((address_space(3)))

// Padded LDS row stride: 132 mod 64 = 4 -> the 16 rows land in 16 distinct
// banks for the fragment reads (128 would alias all rows to the same bank).
#define LDS_STRIDE 132

// ---------------------------------------------------------------------------
// Self-loop GEMM:  C[m, n] = sum_k A[idx(m), k] * W[k, n] + bias[n],  K = 128.
// Block = 8 waves; the block stages one 16x128 A tile in LDS (async DMA,
// double-buffered) and each wave computes one 16x16 output tile for its own
// 16-wide column strip with 32 x V_WMMA_F32_16X16X4_F32, reading A fragments
// from LDS (ds_load_b64) and holding its B strip in 64 VGPRs for the whole
// grid-stride row loop.
//
// VGPR layouts per CDNA5 ISA 7.12.2:
//   A 16x4 f32 : lanes 0-15 -> M=lane,    K={0,1}; lanes 16-31 -> M=lane-16, K={2,3}
//   B 4x16 f32 : lanes 0-15 -> N=lane,    K={0,1}; lanes 16-31 -> N=lane-16, K={2,3}
//   C/D 16x16  : VGPR r: lanes 0-15 -> M=r, N=lane; lanes 16-31 -> M=r+8, N=lane-16
// ---------------------------------------------------------------------------
__global__ __launch_bounds__(256) void kgvae_selfloop_gemm(
    const float* __restrict__ A, const int* __restrict__ nid,
    const float* __restrict__ W, const float* __restrict__ bias,
    float* __restrict__ C, int rowTiles, int nOut)
{
  __shared__ __align__(16) float sA[2][16 * LDS_STRIDE];

  const int lane = threadIdx.x & 31;
  const int wave = threadIdx.x >> 5;
  const int n0   = (blockIdx.y * 8 + wave) * 16;   // this wave's column strip
  const int col  = lane & 15;
  const int kOff = (lane >> 4) << 1;               // 0 or 2
  const int mOff = (lane >> 4) << 3;               // 0 or 8 (C/D row split)

  // Preload B fragments for this strip: 32 k-steps x 2 floats per lane.
  v2f bfrag[32];
#pragma unroll
  for (int ks = 0; ks < 32; ++ks) {
    const int k = ks * 4 + kOff;
    v2f b;
    b.x = W[(size_t)(k + 0) * nOut + n0 + col];
    b.y = W[(size_t)(k + 1) * nOut + n0 + col];
    bfrag[ks] = b;
  }
  const float bv = bias[n0 + col];

  // Stage one 16x128 tile (512 float4s; 2 per thread) into LDS buffer b.
  auto stageTile = [&](int rt, int b) {
    const int t = threadIdx.x;
#pragma unroll
    for (int j = 0; j < 2; ++j) {
      const int f   = t + j * 256;
      const int row = f >> 5;                      // 0..15
      const int c4  = f & 31;                      // 0..31 (float4 column)
      int arow = rt * 16 + row;
      if (nid) arow = nid[arow];                   // layer-1 embedding gather
      const float* gp = A + (size_t)arow * 128 + c4 * 4;
      float*       lp = &sA[b][row * LDS_STRIDE + c4 * 4];
#if defined(USE_ASYNC_LDS)
      __builtin_amdgcn_global_load_async_to_lds_b128(
          (AS1 v4i*)(size_t)gp,
          (AS3 v4i*)(unsigned)(size_t)lp, 0, 0);
#else
      *(float4*)lp = *(const float4*)gp;
#endif
    }
  };

  int rt  = blockIdx.x;                            // every block has >=1 tile
  int cur = 0;
  stageTile(rt, 0);

  for (; rt < rowTiles; rt += gridDim.x) {
    const int  nxt     = rt + (int)gridDim.x;
    const bool hasNext = nxt < rowTiles;
    if (hasNext) stageTile(nxt, cur ^ 1);          // overlap DMA with WMMAs
#if defined(USE_ASYNC_LDS)
    // Each wave has exactly 2 async loads per stage and they retire in
    // order: allowing 2 outstanding leaves only the *next* tile in flight.
    if (hasNext) __builtin_amdgcn_s_wait_asynccnt(2);
    else         __builtin_amdgcn_s_wait_asynccnt(0);
#endif
    __syncthreads();                               // tile `cur` visible to all

    const float* sbuf = &sA[cur][0];
    const float* Ap   = sbuf + (lane & 15) * LDS_STRIDE + kOff;
    v8f acc = {};
#pragma unroll
    for (int ks = 0; ks < 32; ++ks) {
      const v2f a = *(const v2f*)(Ap + ks * 4);    // ds_load_b64, conflict-free
      acc = __builtin_amdgcn_wmma_f32_16x16x4_f32(
          /*neg_a=*/false, a, /*neg_b=*/false, bfrag[ks],
          /*c_mod=*/(short)0, acc, /*reuse_a=*/false, /*reuse_b=*/false);
    }

    const int m0 = rt * 16;
    float* Cp = C + (size_t)(m0 + mOff) * nOut + n0 + col;
#pragma unroll
    for (int r = 0; r < 8; ++r)
      Cp[(size_t)r * nOut] = acc[r] + bv;

    __syncthreads();                               // done reading `cur` before
    cur ^= 1;                                      // it is restaged next iter
  }
}

// ---------------------------------------------------------------------------
// Edge block-diagonal messages + scatter-add.
// One wave per edge, lane b = block index (B=32, SI=4, SO=4 or 8):
//   msg[b*SO+o] = (h[src_e, b*4:(b+1)*4] . W[etype_e, b, :, o]) * norm_e
//   atomicAdd(agg[dst_e, b*SO+o], msg)
// Lane b reads h[s*128 + 4b ..] as one float4 -> a single coalesced 512B
// burst per edge; weight gather is coalesced and L2-resident (409KB/819KB).
// ---------------------------------------------------------------------------
template<int SO>
__global__ __launch_bounds__(256) void kgvae_edge(
    const float* __restrict__ h, const int* __restrict__ nid,
    const int* __restrict__ src, const int* __restrict__ dstv,
    const int* __restrict__ et, const float* __restrict__ norm,
    const float* __restrict__ Wrel, float* __restrict__ agg, int nE)
{
  const int tid = blockIdx.x * 256 + threadIdx.x;
  const int e = tid >> 5;
  const int b = tid & 31;
  if (e >= nE) return;

  int s = src[e];
  if (nid) s = nid[s];                   // layer-1 gathers emb[node_ids[src]]
  const int d   = dstv[e];
  const int r   = et[e];
  const float nm = norm[e];

  const float4 x  = *(const float4*)(h + (size_t)s * 128 + b * 4);
  const float* Wb = Wrel + ((size_t)r * 32 + b) * (4 * SO);
  float* out      = agg + (size_t)d * (32 * SO) + b * SO;

#pragma unroll
  for (int o = 0; o < SO; ++o) {
    const float m = (x.x * Wb[0 * SO + o] + x.y * Wb[1 * SO + o] +
                     x.z * Wb[2 * SO + o] + x.w * Wb[3 * SO + o]) * nm;
    atomicAdd(out + o, m);
  }
}

// In-place ReLU over float4s.
__global__ __launch_bounds__(256) void kgvae_relu(float4* __restrict__ x, int n4)
{
  const int i = blockIdx.x * 256 + threadIdx.x;
  if (i >= n4) return;
  float4 v = x[i];
  v.x = fmaxf(v.x, 0.f); v.y = fmaxf(v.y, 0.f);
  v.z = fmaxf(v.z, 0.f); v.w = fmaxf(v.w, 0.f);
  x[i] = v;
}

// z = m + sqrt(softplus(hp) + 1e-8) * eps  (gaussian split + reparam sample)
__global__ __launch_bounds__(256) void kgvae_finalize(
    const float* __restrict__ agg2, const float* __restrict__ eps,
    float* __restrict__ z, int total /* N*128 */)
{
  const int i = blockIdx.x * 256 + threadIdx.x;
  if (i >= total) return;
  const int row = i >> 7, c = i & 127;
  const float m  = agg2[(size_t)row * 256 + c];
  const float hp = agg2[(size_t)row * 256 + 128 + c];
  const float sp = fmaxf(hp, 0.f) + log1pf(expf(-fabsf(hp)));  // stable softplus
  z[i] = m + sqrtf(sp + 1e-8f) * eps[i];
}

static inline int igmin(int a, int b) { return a < b ? a : b; }

extern "C" void kernel_launch(void* const* d_in, const int* in_sizes, int n_in,
                              void* d_out, int out_size, void* d_ws, size_t ws_size,
                              hipStream_t stream) {
  // setup_inputs() order
  const int*   node_ids = (const int*)  d_in[0];
  const int*   src      = (const int*)  d_in[1];
  const int*   dst      = (const int*)  d_in[2];
  const int*   etypes   = (const int*)  d_in[3];
  const float* norm     = (const float*)d_in[4];
  const float* emb      = (const float*)d_in[5];
  const float* W1       = (const float*)d_in[6];
  const float* loop_w1  = (const float*)d_in[7];
  const float* bias1    = (const float*)d_in[8];
  const float* W2       = (const float*)d_in[9];
  const float* loop_w2  = (const float*)d_in[10];
  const float* bias2    = (const float*)d_in[11];
  const float* eps      = (const float*)d_in[12];
  float*       z        = (float*)d_out;

  const int N_ = in_sizes[0];   // 100000
  const int E_ = in_sizes[1];   // 300000

  // Workspace: agg1/h1 (N*128 f32) then agg2 (N*256 f32).  The GEMM kernels
  // fully overwrite their agg buffer (bias + self-loop), so no memset needed
  // before the edge atomics accumulate on top.
  float* agg1 = (float*)d_ws;
  float* agg2 = agg1 + (size_t)N_ * 128;

  const int rowTiles = N_ / 16;                 // 6250 (N divisible by 16)
  const dim3 blk(256);
  const int gemmBlocksX = igmin(rowTiles, 640); // grid-stride amortizes B preload
  const int edgeBlocks  = (E_ * 32 + 255) / 256;

  // ---- Layer 1 ----
  kgvae_selfloop_gemm<<<dim3(gemmBlocksX, 1), blk, 0, stream>>>(
      emb, node_ids, loop_w1, bias1, agg1, rowTiles, 128);
  kgvae_edge<4><<<edgeBlocks, blk, 0, stream>>>(
      emb, node_ids, src, dst, etypes, norm, W1, agg1, E_);
  kgvae_relu<<<((N_ * 32) + 255) / 256, blk, 0, stream>>>(
      (float4*)agg1, N_ * 32);                  // N*128 floats = N*32 float4

  // ---- Layer 2 ----
  kgvae_selfloop_gemm<<<dim3(gemmBlocksX, 2), blk, 0, stream>>>(
      agg1, nullptr, loop_w2, bias2, agg2, rowTiles, 256);
  kgvae_edge<8><<<edgeBlocks, blk, 0, stream>>>(
      agg1, nullptr, src, dst, etypes, norm, W2, agg2, E_);

  // ---- Gaussian split + reparameterized sample ----
  kgvae_finalize<<<((N_ * 128) + 255) / 256, blk, 0, stream>>>(
      agg2, eps, z, N_ * 128);
}